// QRNN_89799176225061
// MI455X (gfx1250) — compile-verified
//
#include <hip/hip_runtime.h>
#include <math.h>

typedef __attribute__((ext_vector_type(2))) float v2f;
typedef __attribute__((ext_vector_type(8))) float v8f;

#define NQ      11
#define DIM     2048           // 2^11 amplitudes
#define NBATCH  128
#define NSTEP   48
#define NVIS    6
#define NHID    5
#define NFEAT   60
#define TPB     256            // 8 waves (wave32)
#define APT     (DIM / TPB)    // 8 amplitudes per thread
#define TWO_PI_F 6.28318530717958647692f

__device__ __forceinline__ float2 cmul(float2 a, float2 b) {
    return make_float2(a.x * b.x - a.y * b.y, a.x * b.y + a.y * b.x);
}

// wave32 XOR-butterfly add via ds_swizzle (group-of-32 encoding: and=0x1f, xor=MASK)
template<int MASK>
__device__ __forceinline__ float swz_add(float v) {
    int s = __builtin_amdgcn_ds_swizzle(__float_as_int(v), (MASK << 10) | 0x1f);
    return v + __int_as_float(s);
}
__device__ __forceinline__ float wave_sum32(float v) {
    v = swz_add<16>(v);
    v = swz_add<8>(v);
    v = swz_add<4>(v);
    v = swz_add<2>(v);
    v = swz_add<1>(v);
    return v;
}

// ---------------------------------------------------------------------------
// Kernel 1: pooled/min-max preprocess for ALL (b,t) pairs, hoisted out of the
// scan (it does not depend on hidden).  pooled = X(6144x60) @ W(60x6) done
// with v_wmma_f32_16x16x4_f32, one wave per 16-row tile, 15 K-chunks.
// ---------------------------------------------------------------------------
__global__ void __launch_bounds__(32)
pool_angles_kernel(const float* __restrict__ x, float* __restrict__ angV) {
    const int row0 = blockIdx.x * 16;     // row = b*48 + t, 6144 rows total
    const int L    = threadIdx.x;         // 0..31
    const int m    = L & 15;              // M (and N) index for this lane
    const bool hi  = (L >= 16);           // lanes 16..31 carry K = 2,3 of each chunk

    __shared__ float pool[16][16];

    v8f acc = {};
    const float* rowp = x + (row0 + m) * NFEAT;
    #pragma unroll
    for (int kc = 0; kc < 15; ++kc) {
        const int k0 = kc * 4 + (hi ? 2 : 0);
        const int k1 = k0 + 1;
        v2f a, b;
        a.x = rowp[k0];                               // A[m][k0]
        a.y = rowp[k1];                               // A[m][k1]
        b.x = (m < NVIS && (k0 / 10) == m) ? 0.1f : 0.0f;  // B[k0][n]
        b.y = (m < NVIS && (k1 / 10) == m) ? 0.1f : 0.0f;  // B[k1][n]
        acc = __builtin_amdgcn_wmma_f32_16x16x4_f32(
            false, a, false, b, (short)0, acc, false, false);
    }
    // D layout: vgpr r -> row (r + hi*8), col = lane&15
    #pragma unroll
    for (int r = 0; r < 8; ++r)
        pool[r + (hi ? 8 : 0)][m] = acc[r];
    __syncthreads();

    if (L < 16) {
        float p[NVIS];
        float mn = 1e30f, mx = -1e30f;
        #pragma unroll
        for (int v = 0; v < NVIS; ++v) {
            p[v] = pool[L][v];
            mn = fminf(mn, p[v]);
            mx = fmaxf(mx, p[v]);
        }
        const float inv = TWO_PI_F / (mx - mn + 1e-8f);
        float* o = angV + (row0 + L) * NVIS;
        #pragma unroll
        for (int v = 0; v < NVIS; ++v) o[v] = (p[v] - mn) * inv;
    }
}

// ---------------------------------------------------------------------------
// Kernel 2: the 48-step scan.  One block per batch sample; 2048-amp complex
// state lives in LDS; 74 per-step gates pre-fused into 14 4x4 complex gates.
// ---------------------------------------------------------------------------

// U = RZ(tz) * RY(ty) * RX(tx), row-major 2x2 complex
__device__ __forceinline__ void rot_u(float tx, float ty, float tz, float2* U) {
    float cx = cosf(0.5f * tx), sx = sinf(0.5f * tx);
    float cy = cosf(0.5f * ty), sy = sinf(0.5f * ty);
    float cz = cosf(0.5f * tz), sz = sinf(0.5f * tz);
    // M = RY*RX
    float2 M00 = make_float2( cy * cx,  sy * sx);
    float2 M01 = make_float2(-sy * cx, -cy * sx);
    float2 M10 = make_float2( sy * cx, -cy * sx);
    float2 M11 = make_float2( cy * cx, -sy * sx);
    float2 z0 = make_float2(cz, -sz), z1 = make_float2(cz, sz);
    U[0] = cmul(z0, M00); U[1] = cmul(z0, M01);
    U[2] = cmul(z1, M10); U[3] = cmul(z1, M11);
}

#define CMA(acc, g, v)                                   \
    do {                                                 \
        (acc).x += (g).x * (v).x - (g).y * (v).y;        \
        (acc).y += (g).x * (v).y + (g).y * (v).x;        \
    } while (0)

// Apply a 4x4 complex gate on qubit pair with bit positions (BB+1, BB).
template<int BB>
__device__ __forceinline__ void apply_gate2q(float2* state, const float2* G, int tid) {
    #pragma unroll
    for (int gi = 0; gi < 2; ++gi) {
        const int grp = tid + gi * TPB;  // 0..511
        const int i0  = ((grp >> BB) << (BB + 2)) | (grp & ((1 << BB) - 1));
        const int db  = 1 << BB;
        float2 v0 = state[i0];
        float2 v1 = state[i0 + db];
        float2 v2 = state[i0 + 2 * db];
        float2 v3 = state[i0 + 3 * db];
        float2 n0 = make_float2(0.f, 0.f), n1 = n0, n2 = n0, n3 = n0;
        CMA(n0, G[0],  v0); CMA(n0, G[1],  v1); CMA(n0, G[2],  v2); CMA(n0, G[3],  v3);
        CMA(n1, G[4],  v0); CMA(n1, G[5],  v1); CMA(n1, G[6],  v2); CMA(n1, G[7],  v3);
        CMA(n2, G[8],  v0); CMA(n2, G[9],  v1); CMA(n2, G[10], v2); CMA(n2, G[11], v3);
        CMA(n3, G[12], v0); CMA(n3, G[13], v1); CMA(n3, G[14], v2); CMA(n3, G[15], v3);
        state[i0]          = n0;
        state[i0 + db]     = n1;
        state[i0 + 2 * db] = n2;
        state[i0 + 3 * db] = n3;
    }
}

__global__ void __launch_bounds__(TPB)
qrnn_scan_kernel(const float* __restrict__ angV,
                 const float* __restrict__ hidden0,
                 const float* __restrict__ params,
                 float* __restrict__ out) {
    __shared__ float2 state[DIM];        // 16 KB
    __shared__ float2 sGate[14][16];     // fused 4x4 complex gates
    __shared__ float2 cs[NQ];            // (cos,sin) of half-angles
    __shared__ float  sHid[NHID];
    __shared__ float  sRed[8][12];
    __shared__ float  sZ[12];

    const int b   = blockIdx.x;
    const int tid = threadIdx.x;

    // ---- one-time gate fusion (params are time/batch invariant) ----
    if (tid < 14) {
        const int l = tid / 7, s = tid % 7;
        const int base = l * 37;
        float2 Ua[4], Ub[4], K[16];
        if (s < 4) {
            if (s == 0) {
                rot_u(params[base + 0], params[base + 1], params[base + 2], Ua);
                rot_u(params[base + 3], params[base + 4], params[base + 5], Ub);
            } else {
                Ua[0] = make_float2(1.f, 0.f); Ua[1] = make_float2(0.f, 0.f);
                Ua[2] = make_float2(0.f, 0.f); Ua[3] = make_float2(1.f, 0.f);
                const int q = s + 1;  // hidden qubit 2,3,4
                rot_u(params[base + 3 * q], params[base + 3 * q + 1],
                      params[base + 3 * q + 2], Ub);
            }
        } else {
            const int qa = 5 + 2 * (s - 4);          // 5,7,9
            const int pa = base + 19 + 3 * (qa - 5);
            const int pb = base + 19 + 3 * (qa - 4);
            rot_u(params[pa], params[pa + 1], params[pa + 2], Ua);
            rot_u(params[pb], params[pb + 1], params[pb + 2], Ub);
        }
        #pragma unroll
        for (int r = 0; r < 4; ++r)
            #pragma unroll
            for (int c = 0; c < 4; ++c)
                K[r * 4 + c] = cmul(Ua[(r >> 1) * 2 + (c >> 1)],
                                    Ub[(r & 1) * 2 + (c & 1)]);
        if (s < 4) {  // premultiply CRX(theta): rows 2,3 mix with (c, -i s)
            const float th = params[base + 15 + s];
            const float cc = cosf(0.5f * th), ss = sinf(0.5f * th);
            #pragma unroll
            for (int c = 0; c < 4; ++c) {
                float2 k2 = K[8 + c], k3 = K[12 + c];
                K[8 + c]  = make_float2(cc * k2.x + ss * k3.y, cc * k2.y - ss * k3.x);
                K[12 + c] = make_float2(ss * k2.y + cc * k3.x, -ss * k2.x + cc * k3.y);
            }
        }
        #pragma unroll
        for (int e = 0; e < 16; ++e) sGate[tid][e] = K[e];
    }
    if (tid < NHID) sHid[tid] = hidden0[b * NHID + tid];
    __syncthreads();

    const float* av = angV + b * NSTEP * NVIS;

    #pragma unroll 1
    for (int t = 0; t < NSTEP; ++t) {
        // ---- angles -> (cos,sin) ----
        if (tid < NQ) {
            const float a = (tid < NHID) ? sHid[tid] : av[t * NVIS + (tid - NHID)];
            cs[tid] = make_float2(cosf(0.5f * a), sinf(0.5f * a));
        }
        __syncthreads();

        // ---- embed product state: amp(i) = prod(mag) * (-i)^popcount(i) ----
        {
            float pre = 1.f;
            #pragma unroll
            for (int q = 0; q < 8; ++q)
                pre *= ((tid >> (7 - q)) & 1) ? cs[q].y : cs[q].x;
            const int kpre = __popc(tid);
            #pragma unroll
            for (int j = 0; j < APT; ++j) {
                float mag = pre;
                mag *= (j & 4) ? cs[8].y  : cs[8].x;
                mag *= (j & 2) ? cs[9].y  : cs[9].x;
                mag *= (j & 1) ? cs[10].y : cs[10].x;
                const int k = (kpre + __popc(j)) & 3;   // (-i)^k
                const float re = (k == 0) ? mag : ((k == 2) ? -mag : 0.f);
                const float im = (k == 1) ? -mag : ((k == 3) ? mag : 0.f);
                state[tid * APT + j] = make_float2(re, im);
            }
        }
        __syncthreads();

        // ---- 14 fused two-qubit gates (qubit pairs are adjacent) ----
        #pragma unroll
        for (int l = 0; l < 2; ++l) {
            apply_gate2q<9>(state, sGate[l * 7 + 0], tid); __syncthreads(); // (q0,q1)
            apply_gate2q<8>(state, sGate[l * 7 + 1], tid); __syncthreads(); // (q1,q2)
            apply_gate2q<7>(state, sGate[l * 7 + 2], tid); __syncthreads(); // (q2,q3)
            apply_gate2q<6>(state, sGate[l * 7 + 3], tid); __syncthreads(); // (q3,q4)
            apply_gate2q<4>(state, sGate[l * 7 + 4], tid); __syncthreads(); // (q5,q6)
            apply_gate2q<2>(state, sGate[l * 7 + 5], tid); __syncthreads(); // (q7,q8)
            apply_gate2q<0>(state, sGate[l * 7 + 6], tid); __syncthreads(); // (q9,q10)
        }

        // ---- readout: z_q = total - 2 * sum_{bit_q=1} |amp|^2 ----
        {
            float tot = 0.f, s8 = 0.f, s9 = 0.f, s10 = 0.f;
            #pragma unroll
            for (int j = 0; j < APT; ++j) {
                const float2 a = state[tid * APT + j];
                const float p = a.x * a.x + a.y * a.y;
                tot += p;
                if (j & 4) s8  += p;
                if (j & 2) s9  += p;
                if (j & 1) s10 += p;
            }
            float vals[12];
            #pragma unroll
            for (int q = 0; q < 8; ++q)
                vals[q] = ((tid >> (7 - q)) & 1) ? tot : 0.f;
            vals[8] = s8; vals[9] = s9; vals[10] = s10; vals[11] = tot;
            #pragma unroll
            for (int q = 0; q < 12; ++q) vals[q] = wave_sum32(vals[q]);
            const int wave = tid >> 5, lane = tid & 31;
            if (lane == 0) {
                #pragma unroll
                for (int q = 0; q < 12; ++q) sRed[wave][q] = vals[q];
            }
        }
        __syncthreads();
        if (tid < 12) {
            float ssum = 0.f;
            #pragma unroll
            for (int w = 0; w < 8; ++w) ssum += sRed[w][tid];
            sZ[tid] = ssum;
        }
        __syncthreads();
        if (tid < NQ) {
            const float z = sZ[11] - 2.f * sZ[tid];
            if (tid < NHID) {
                sHid[tid] = z;
                if (t == NSTEP - 1)
                    out[NBATCH * NSTEP * NVIS + b * NHID + tid] = z;  // hidden_final
            } else {
                out[b * NSTEP * NVIS + t * NVIS + (tid - NHID)] = z;  // outputs
            }
        }
        __syncthreads();
    }
}

// ---------------------------------------------------------------------------
extern "C" void kernel_launch(void* const* d_in, const int* in_sizes, int n_in,
                              void* d_out, int out_size, void* d_ws, size_t ws_size,
                              hipStream_t stream) {
    const float* x      = (const float*)d_in[0];   // (128, 48, 60)
    const float* hidden = (const float*)d_in[1];   // (128, 5)
    const float* params = (const float*)d_in[2];   // (74,)
    float* out  = (float*)d_out;                   // 128*48*6 outputs + 128*5 hidden
    float* angV = (float*)d_ws;                    // 128*48*6 floats = 147456 B scratch

    // 6144 rows / 16 rows-per-wave-tile = 384 single-wave blocks
    pool_angles_kernel<<<NBATCH * NSTEP / 16, 32, 0, stream>>>(x, angV);
    // one block per batch sample, 48-step scan in-kernel
    qrnn_scan_kernel<<<NBATCH, TPB, 0, stream>>>(angV, hidden, params, out);
}